// GNN_node_44890998178249
// MI455X (gfx1250) — compile-verified
//
#include <hip/hip_runtime.h>

#define Nn 50000
#define Ee 800000
#define Dd 128
#define Ll 3
#define NTILES (Nn / 16)   // 3125 row tiles of 16

typedef float v2f __attribute__((ext_vector_type(2)));
typedef float v8f __attribute__((ext_vector_type(8)));
typedef int   v4i __attribute__((ext_vector_type(4)));

#if defined(__has_builtin)
#if __has_builtin(__builtin_amdgcn_global_load_async_to_lds_b128)
#define USE_ASYNC_LDS 1
#endif
#endif

// ---------------- zero ----------------
__global__ void zero_kernel(float* __restrict__ p, int n) {
    int i = blockIdx.x * blockDim.x + threadIdx.x;
    if (i < n) p[i] = 0.0f;
}

// ---------------- AtomEncoder: h[n,:] = sum_c atom_emb[c, x[n,c], :] ----------------
// 1 wave per node, lane covers 4 consecutive floats of the 128-wide row.
__global__ void atom_encode_kernel(const int* __restrict__ x,
                                   const float* __restrict__ emb, // [9][64][128]
                                   float* __restrict__ h) {
    // node index is wave-uniform: force scalar so index loads become s_load
    int n = __builtin_amdgcn_readfirstlane(blockIdx.x * 8 + (threadIdx.x >> 5));
    int lane = threadIdx.x & 31;
    if (n >= Nn) return;
    int d4 = lane * 4;
    float4 acc = make_float4(0.f, 0.f, 0.f, 0.f);
    #pragma unroll
    for (int c = 0; c < 9; ++c) {
        int v = x[n * 9 + c];
        float4 t = *(const float4*)(emb + ((size_t)(c * 64 + v)) * Dd + d4);
        acc.x += t.x; acc.y += t.y; acc.z += t.z; acc.w += t.w;
    }
    *(float4*)(h + (size_t)n * Dd + d4) = acc;
}

// ---------------- Edge pass: m = relu(h[src] + bond-embed); atomic scatter to agg[dst] ----------------
// 1 wave per edge; lane covers 4 floats. agg must be pre-zeroed.
__global__ void edge_scatter_kernel(const float* __restrict__ h,
                                    const int* __restrict__ src,
                                    const int* __restrict__ dst,
                                    const int* __restrict__ ea,     // [E][3]
                                    const float* __restrict__ bond, // layer slice [3][8][128]
                                    float* __restrict__ agg) {
    // edge index is wave-uniform -> scalar index loads
    int e = __builtin_amdgcn_readfirstlane(blockIdx.x * 8 + (threadIdx.x >> 5));
    int lane = threadIdx.x & 31;
    if (e >= Ee) return;
    int s  = src[e];
    int t  = dst[e];
    int e0 = ea[e * 3 + 0];
    int e1 = ea[e * 3 + 1];
    int e2 = ea[e * 3 + 2];
    int d4 = lane * 4;
    float4 hv = *(const float4*)(h + (size_t)s * Dd + d4);
    float4 b0 = *(const float4*)(bond + (size_t)(0 * 8 + e0) * Dd + d4);
    float4 b1 = *(const float4*)(bond + (size_t)(1 * 8 + e1) * Dd + d4);
    float4 b2 = *(const float4*)(bond + (size_t)(2 * 8 + e2) * Dd + d4);
    float m0 = fmaxf(hv.x + b0.x + b1.x + b2.x, 0.f);
    float m1 = fmaxf(hv.y + b0.y + b1.y + b2.y, 0.f);
    float m2 = fmaxf(hv.z + b0.z + b1.z + b2.z, 0.f);
    float m3 = fmaxf(hv.w + b0.w + b1.w + b2.w, 0.f);
    float* ap = agg + (size_t)t * Dd + d4;
    atomicAdd(ap + 0, m0);
    atomicAdd(ap + 1, m1);
    atomicAdd(ap + 2, m2);
    atomicAdd(ap + 3, m3);
}

// ---------------- WMMA GEMM: C = A @ W + bias, fused BN statistics ----------------
// COMBINE=1: A[n,:] = (1+eps)*H[n,:] + G[n,:]   (GIN combine); COMBINE=0: A = H.
// Block: 256 threads = 8 waves. Wave w owns column tile [16w,16w+16); block owns
// FOUR 16-row tiles (64 rows): one B fragment feeds 4 WMMAs per k-step.
// f32 WMMA 16x16x4 fragment layout per ISA:
//   A: lane (m = lane&15) holds K pair at k + 2*(lane>=16)  -> contiguous float2
//   B: lane (n = lane&15) holds K pair at k + 2*(lane>=16)  (stride-D apart)
//   C: VGPR r -> M = r + 8*(lane>=16), N = lane&15
template <int COMBINE>
__global__ void gemm_bn_stats_kernel(const float* __restrict__ H,
                                     const float* __restrict__ G,
                                     const float* __restrict__ W,     // [128][128] (k, col)
                                     const float* __restrict__ bias,  // [128]
                                     const float* __restrict__ epsp,  // &gin_eps[l]
                                     float* __restrict__ C,           // [N][128]
                                     float* __restrict__ stats) {     // [256]: colsum, colsumsq
    __shared__ float sW[Dd * Dd]; // 64 KB of the WGP's 320 KB LDS
    int tid = threadIdx.x;

    // ---- stage W into LDS: async copy path (ASYNCcnt) with sync fallback ----
#ifdef USE_ASYNC_LDS
    {
        __attribute__((address_space(1))) v4i* gw =
            (__attribute__((address_space(1))) v4i*)W;
        __attribute__((address_space(3))) v4i* lw =
            (__attribute__((address_space(3))) v4i*)sW;
        for (int i = tid; i < Dd * Dd / 4; i += 256) {
            __builtin_amdgcn_global_load_async_to_lds_b128(gw + i, lw + i, 0, 0);
        }
#if __has_builtin(__builtin_amdgcn_s_wait_asynccnt)
        __builtin_amdgcn_s_wait_asynccnt(0);
#else
        asm volatile("s_wait_asynccnt 0x0" ::: "memory");
#endif
    }
#else
    for (int i = tid; i < Dd * Dd / 4; i += 256) {
        ((float4*)sW)[i] = ((const float4*)W)[i];
    }
#endif
    __syncthreads();

    int wave  = tid >> 5;
    int lane  = tid & 31;
    int col   = wave * 16 + (lane & 15);
    int rlane = lane & 15;
    int koff  = (lane >> 4) << 1; // 0 or 2
    int tile0 = blockIdx.x * 4;   // first of 4 row tiles handled by this block
    float epsv = 1.0f + epsp[0];

    const float* Arow[4];
    const float* Grow[4];
    #pragma unroll
    for (int t = 0; t < 4; ++t) {
        int row = (tile0 + t) * 16 + rlane;
        if (row > Nn - 1) row = Nn - 1;   // clamp: keeps loads in-bounds, EXEC all-1s
        Arow[t] = H + (size_t)row * Dd;
        Grow[t] = G + (size_t)row * Dd;
    }

    v8f acc[4] = {};
    #pragma unroll 2
    for (int k = 0; k < Dd; k += 4) {
        int ki = k + koff;
        v2f b;
        b[0] = sW[ki * Dd + col];
        b[1] = sW[(ki + 1) * Dd + col];
        #pragma unroll
        for (int t = 0; t < 4; ++t) {
            v2f a;
            if (COMBINE) {
                a[0] = epsv * Arow[t][ki]     + Grow[t][ki];
                a[1] = epsv * Arow[t][ki + 1] + Grow[t][ki + 1];
            } else {
                a[0] = Arow[t][ki];
                a[1] = Arow[t][ki + 1];
            }
            acc[t] = __builtin_amdgcn_wmma_f32_16x16x4_f32(false, a, false, b,
                                                           (short)0, acc[t], false, false);
        }
    }

    // epilogue: bias + per-column partial stats + store (guard invalid tiles)
    float bv = bias[col];
    #pragma unroll
    for (int t = 0; t < 4; ++t) {
        int tile = tile0 + t;
        if (tile < NTILES) {
            int mb = tile * 16 + ((lane >> 4) << 3);
            float s = 0.f, sq = 0.f;
            #pragma unroll
            for (int r = 0; r < 8; ++r) {
                float v = acc[t][r] + bv;
                s  += v;
                sq += v * v;
                C[(size_t)(mb + r) * Dd + col] = v;
            }
            atomicAdd(&stats[col], s);
            atomicAdd(&stats[Dd + col], sq);
        }
    }
}

// ---------------- BN apply (+optional ReLU): y = g*(x-mean)*rsqrt(var+eps)+b ----------------
__global__ void bn_apply_kernel(const float* __restrict__ Z,
                                float* __restrict__ Out,
                                const float* __restrict__ stats,
                                const float* __restrict__ gamma,
                                const float* __restrict__ beta,
                                int do_relu) {
    int idx = blockIdx.x * blockDim.x + threadIdx.x;
    if (idx >= Nn * Dd) return;
    int col = idx & (Dd - 1);
    const float invN = 1.0f / (float)Nn;
    float mean = stats[col] * invN;
    float var  = stats[Dd + col] * invN - mean * mean;
    float inv  = rsqrtf(var + 1e-5f);
    float v = gamma[col] * (Z[idx] - mean) * inv + beta[col];
    if (do_relu) v = fmaxf(v, 0.f);
    Out[idx] = v;
}

extern "C" void kernel_launch(void* const* d_in, const int* in_sizes, int n_in,
                              void* d_out, int out_size, void* d_ws, size_t ws_size,
                              hipStream_t stream) {
    (void)in_sizes; (void)n_in; (void)out_size; (void)ws_size;
    const int*   x         = (const int*)  d_in[0];
    const int*   edge_attr = (const int*)  d_in[1];
    const int*   src       = (const int*)  d_in[2];
    const int*   dst       = (const int*)  d_in[3];
    const float* atom_emb  = (const float*)d_in[4];
    const float* bond_emb  = (const float*)d_in[5];  // [L][3][8][128]
    const float* W1        = (const float*)d_in[6];  // [L][128][128]
    const float* b1        = (const float*)d_in[7];
    const float* g1        = (const float*)d_in[8];
    const float* be1       = (const float*)d_in[9];
    const float* W2        = (const float*)d_in[10];
    const float* b2        = (const float*)d_in[11];
    const float* gin_eps   = (const float*)d_in[12]; // [L]
    const float* og        = (const float*)d_in[13];
    const float* ob        = (const float*)d_in[14];

    float* h     = (float*)d_out;                 // [N][128], evolves per layer
    float* agg   = (float*)d_ws;                  // [N][128]
    float* z     = agg + (size_t)Nn * Dd;         // [N][128]
    float* stats = z + (size_t)Nn * Dd;           // [256]

    const int ND = Nn * Dd;
    const int gemm_blocks = (NTILES + 3) / 4;     // 4 row tiles per block

    // AtomEncoder
    atom_encode_kernel<<<Nn / 8, 256, 0, stream>>>(x, atom_emb, h);

    for (int l = 0; l < Ll; ++l) {
        // zero agg + stats
        zero_kernel<<<(ND + 255) / 256, 256, 0, stream>>>(agg, ND);
        zero_kernel<<<1, 256, 0, stream>>>(stats, 256);

        // edge gather/embed/relu/scatter-sum
        edge_scatter_kernel<<<Ee / 8, 256, 0, stream>>>(
            h, src, dst, edge_attr, bond_emb + (size_t)l * 3 * 8 * Dd, agg);

        // z = ((1+eps)*h + agg) @ W1 + b1, with fused column stats
        gemm_bn_stats_kernel<1><<<gemm_blocks, 256, 0, stream>>>(
            h, agg, W1 + (size_t)l * Dd * Dd, b1 + l * Dd, gin_eps + l, z, stats);

        // z = relu(BN(z))   (in place)
        bn_apply_kernel<<<(ND + 255) / 256, 256, 0, stream>>>(
            z, z, stats, g1 + l * Dd, be1 + l * Dd, 1);

        // second GEMM: agg reused as z2 output
        zero_kernel<<<1, 256, 0, stream>>>(stats, 256);
        gemm_bn_stats_kernel<0><<<gemm_blocks, 256, 0, stream>>>(
            z, z /*unused*/, W2 + (size_t)l * Dd * Dd, b2 + l * Dd, gin_eps + l, agg, stats);

        // h = BN(z2) (+relu except last layer)
        bn_apply_kernel<<<(ND + 255) / 256, 256, 0, stream>>>(
            agg, h, stats, og + l * Dd, ob + l * Dd, (l < Ll - 1) ? 1 : 0);
    }
}